// acm_hGCN_21534966022323
// MI455X (gfx1250) — compile-verified
//
#include <hip/hip_runtime.h>
#include <hip/hip_bf16.h>
#include <cstdint>
#include <cstddef>

#define N_NODES 100000
#define N_EDGES 1600000
#define D       128

typedef __attribute__((ext_vector_type(2))) float v2f;
typedef __attribute__((ext_vector_type(8))) float v8f;

__device__ __forceinline__ void atomAddF(float* p, float v) {
  (void)__hip_atomic_fetch_add(p, v, __ATOMIC_RELAXED, __HIP_MEMORY_SCOPE_AGENT);
}

__device__ __forceinline__ float eluf(float x) {
  return x > 0.f ? x : expm1f(x);
}

// ---------------------------------------------------------------------------
// Degree counting: one thread per edge, hardware f32 atomics.
// ---------------------------------------------------------------------------
__global__ void degree_kernel(const int* __restrict__ src, const int* __restrict__ dst,
                              float* __restrict__ degO, float* __restrict__ degI, int E) {
  int i = blockIdx.x * blockDim.x + threadIdx.x;
  if (i < E) {
    atomAddF(degO + src[i], 1.f);
    atomAddF(degI + dst[i], 1.f);
  }
}

// In-place deg -> deg^{-1/2}, 0-degree -> 0 (matches DGL norm='both').
__global__ void isqrt_kernel(float* __restrict__ d, int n) {
  int i = blockIdx.x * blockDim.x + threadIdx.x;
  if (i < n) {
    float x = d[i];
    d[i] = (x == 0.f) ? 0.f : rsqrtf(x);
  }
}

// ---------------------------------------------------------------------------
// fp32 WMMA GEMM: Y[M x 128] = X[M x K] @ W[K x 128] + bias (optional ELU).
// One wave per 16x16 tile; 8 waves/block span the full N=128.
// A-tile layout (16x4 f32): lanes 0-15 hold K=0,1; lanes 16-31 hold K=2,3.
// C/D layout (16x16 f32):   VGPR i -> row m0 + i + 8*(lane>=16), col n0+(lane&15).
// ---------------------------------------------------------------------------
template <bool ELU>
__global__ void gemm_wmma_f32(const float* __restrict__ X, const float* __restrict__ W,
                              const float* __restrict__ bias, float* __restrict__ Y,
                              int M, int K) {
  const int lane = threadIdx.x & 31;
  const int wave = threadIdx.x >> 5;           // 0..7 -> N tile
  const int m0   = blockIdx.x * 16;
  const int n0   = wave * 16;
  const int l15  = lane & 15;
  const int row  = m0 + l15;
  const int col  = n0 + l15;
  const int koff = (lane >> 4) << 1;           // 0 for lanes 0-15, 2 for lanes 16-31

  v8f acc = {};
  const float* xrow = X + (size_t)row * K;
  #pragma unroll 4
  for (int k = 0; k < K; k += 4) {
    v2f a = *(const v2f*)(xrow + k + koff);    // 8B-aligned: K and koff even
    v2f b;
    b.x = W[(size_t)(k + koff    ) * D + col];
    b.y = W[(size_t)(k + koff + 1) * D + col];
    acc = __builtin_amdgcn_wmma_f32_16x16x4_f32(
        /*neg_a=*/false, a, /*neg_b=*/false, b,
        /*c_mod=*/(short)0, acc, /*reuse_a=*/false, /*reuse_b=*/false);
  }

  const float bc    = bias[col];
  const int   rbase = m0 + ((lane >> 4) << 3); // +0 or +8
  #pragma unroll
  for (int i = 0; i < 8; ++i) {
    float v = acc[i] + bc;
    if (ELU) v = eluf(v);
    Y[(size_t)(rbase + i) * D + col] = v;
  }
}

// ---------------------------------------------------------------------------
// Y[node,:] = X[node,:] * s[node]   (float4 per lane, 32 lanes per node)
// ---------------------------------------------------------------------------
__global__ void row_scale(const float* __restrict__ X, const float* __restrict__ s,
                          float* __restrict__ Y, int n) {
  int gid  = blockIdx.x * blockDim.x + threadIdx.x;
  int node = gid >> 5;
  int lane = gid & 31;
  if (node >= n) return;
  float  sc = s[node];
  float4 v  = *(const float4*)(X + (size_t)node * D + lane * 4);
  v.x *= sc; v.y *= sc; v.z *= sc; v.w *= sc;
  *(float4*)(Y + (size_t)node * D + lane * 4) = v;
}

// ---------------------------------------------------------------------------
// Y[node,:] = elu(A[node,:] * isq[node] + bias[:])   (layer-0 epilogue)
// ---------------------------------------------------------------------------
__global__ void scale_bias_elu(const float* __restrict__ A, const float* __restrict__ isq,
                               const float* __restrict__ bias, float* __restrict__ Y, int n) {
  int gid  = blockIdx.x * blockDim.x + threadIdx.x;
  int node = gid >> 5;
  int lane = gid & 31;
  if (node >= n) return;
  float  sc = isq[node];
  float4 v  = *(const float4*)(A + (size_t)node * D + lane * 4);
  float4 b  = *(const float4*)(bias + lane * 4);
  float4 o;
  o.x = eluf(v.x * sc + b.x);
  o.y = eluf(v.y * sc + b.y);
  o.z = eluf(v.z * sc + b.z);
  o.w = eluf(v.w * sc + b.w);
  *(float4*)(Y + (size_t)node * D + lane * 4) = o;
}

// ---------------------------------------------------------------------------
// Edge scatter: one wave per edge, float4 per lane.
//   MODE 0: out[dst] += H[src]                    (aggregation, H pre-scaled)
//   MODE 1: out[dst] += H[src] * (e&1 ? 1 : 2)    (final message pass)
// ---------------------------------------------------------------------------
template <int MODE>
__global__ void edge_scatter(const float* __restrict__ H, const int* __restrict__ src,
                             const int* __restrict__ dst, const int* __restrict__ ef,
                             float* __restrict__ out, int E) {
  int gid  = blockIdx.x * blockDim.x + threadIdx.x;
  int e    = gid >> 5;
  int lane = gid & 31;
  if (e >= E) return;

  float s = 1.f;
  if (MODE == 1) s = (ef[e] & 1) ? 1.f : 2.f;   // e_feat in {0,2,4,6} -> x2

  const float4 v = *(const float4*)(H + (size_t)src[e] * D + lane * 4);
  float* o = out + (size_t)dst[e] * D + lane * 4;
  atomAddF(o + 0, v.x * s);
  atomAddF(o + 1, v.y * s);
  atomAddF(o + 2, v.z * s);
  atomAddF(o + 3, v.w * s);
}

// ---------------------------------------------------------------------------
extern "C" void kernel_launch(void* const* d_in, const int* in_sizes, int n_in,
                              void* d_out, int out_size, void* d_ws, size_t ws_size,
                              hipStream_t stream) {
  const float* feat0  = (const float*)d_in[0];
  const float* feat1  = (const float*)d_in[1];
  const float* feat2  = (const float*)d_in[2];
  const float* W0     = (const float*)d_in[3];
  const float* b0     = (const float*)d_in[4];
  const float* W1     = (const float*)d_in[5];
  const float* b1     = (const float*)d_in[6];
  const float* W2     = (const float*)d_in[7];
  const float* b2     = (const float*)d_in[8];
  const float* gcn0_b = (const float*)d_in[9];
  const float* gcn1_W = (const float*)d_in[10];
  const float* gcn1_b = (const float*)d_in[11];
  const int*   e_feat = (const int*)d_in[12];
  const int*   src    = (const int*)d_in[13];
  const int*   dst    = (const int*)d_in[14];
  float*       out    = (float*)d_out;

  const size_t FEAT_BYTES = (size_t)N_NODES * D * sizeof(float);  // 51.2 MB
  float* bufA = (float*)d_ws;
  float* bufB = (float*)((char*)d_ws + FEAT_BYTES);
  float* degO = (float*)((char*)d_ws + 2 * FEAT_BYTES);           // -> out_isqrt
  float* degI = degO + N_NODES;                                   // -> in_isqrt

  const int BLK        = 256;
  const int edgeBlks   = (N_EDGES + BLK - 1) / BLK;               // 1 thread/edge
  const int nodeBlks   = (N_NODES + BLK - 1) / BLK;               // 1 thread/node
  const int nodeVBlks  = (int)(((size_t)N_NODES * 32 + BLK - 1) / BLK);  // 1 wave/node
  const int edgeVBlks  = (int)(((size_t)N_EDGES * 32 + BLK - 1) / BLK);  // 1 wave/edge

  // --- degrees -> inverse-sqrt norms -------------------------------------
  hipMemsetAsync(degO, 0, 2 * (size_t)N_NODES * sizeof(float), stream);
  degree_kernel<<<edgeBlks, BLK, 0, stream>>>(src, dst, degO, degI, N_EDGES);
  isqrt_kernel<<<nodeBlks, BLK, 0, stream>>>(degO, N_NODES);
  isqrt_kernel<<<nodeBlks, BLK, 0, stream>>>(degI, N_NODES);

  // --- per-type input projections (WMMA) into bufA ------------------------
  gemm_wmma_f32<false><<<40000 / 16, BLK, 0, stream>>>(feat0, W0, b0, bufA,                 40000, 256);
  gemm_wmma_f32<false><<<30000 / 16, BLK, 0, stream>>>(feat1, W1, b1, bufA + 40000ull * D,  30000, 128);
  gemm_wmma_f32<false><<<30000 / 16, BLK, 0, stream>>>(feat2, W2, b2, bufA + 70000ull * D,  30000,  64);

  // --- GraphConv layer 0 (weight=False) ----------------------------------
  row_scale<<<nodeVBlks, BLK, 0, stream>>>(bufA, degO, bufB, N_NODES);  // h * out_isqrt
  hipMemsetAsync(bufA, 0, FEAT_BYTES, stream);
  edge_scatter<0><<<edgeVBlks, BLK, 0, stream>>>(bufB, src, dst, nullptr, bufA, N_EDGES);
  scale_bias_elu<<<nodeVBlks, BLK, 0, stream>>>(bufA, degI, gcn0_b, bufA, N_NODES);  // h1

  // --- GraphConv layer 1 (weighted) ---------------------------------------
  row_scale<<<nodeVBlks, BLK, 0, stream>>>(bufA, degO, bufB, N_NODES);  // h1 * out_isqrt
  hipMemsetAsync(bufA, 0, FEAT_BYTES, stream);
  edge_scatter<0><<<edgeVBlks, BLK, 0, stream>>>(bufB, src, dst, nullptr, bufA, N_EDGES);
  row_scale<<<nodeVBlks, BLK, 0, stream>>>(bufA, degI, bufA, N_NODES);  // * in_isqrt
  gemm_wmma_f32<true><<<N_NODES / 16, BLK, 0, stream>>>(bufA, gcn1_W, gcn1_b, bufB, N_NODES, 128); // h2

  // --- final typed message pass -------------------------------------------
  hipMemsetAsync(out, 0, (size_t)out_size * sizeof(float), stream);
  edge_scatter<1><<<edgeVBlks, BLK, 0, stream>>>(bufB, src, dst, e_feat, out, N_EDGES);

  (void)in_sizes; (void)n_in; (void)ws_size;
}